// RasterizePointsXYsBlending_28810640621827
// MI455X (gfx1250) — compile-verified
//
#include <hip/hip_runtime.h>

typedef __attribute__((ext_vector_type(2))) float v2f;
typedef __attribute__((ext_vector_type(4))) float v4f;
typedef __attribute__((ext_vector_type(8))) float v8f;

#define IMG   128
#define NPTS  2048
#define NCH   64
#define KSEL  8
#define TILE  16
#define MAXC  2048
#define BIGZ  1e10f

__launch_bounds__(256)
__global__ void raster_tile_wmma(const float* __restrict__ pts3D,  // [B,P,3]
                                 const float* __restrict__ src,    // [B,C,P]
                                 float* __restrict__ out)          // [B,C,H,W]
{
    const int b    = blockIdx.z;
    const int tw   = blockIdx.x;
    const int th   = blockIdx.y;
    const int tid  = threadIdx.x;        // 0..255
    const int lane = tid & 31;
    const int wv   = tid >> 5;           // wave id 0..7 (wave32)

    const float r_ndc  = 2.0f * 1.5f / (float)IMG;
    const float r2     = r_ndc * r_ndc;
    const float inv_r2 = 1.0f / r2;

    __shared__ int   s_count;
    __shared__ int   s_pidx[MAXC];
    __shared__ float s_px[MAXC];
    __shared__ float s_py[MAXC];
    __shared__ float s_pz[MAXC];
    __shared__ float s_B[256 * 16];      // weight chunk, TRANSPOSED: [pixel][kk]
    __shared__ float s_A[NCH * 16];      // feature chunk: [c][kk]

    const float* srcb = src + (size_t)b * NCH * NPTS;
    __builtin_prefetch(srcb, 0, 1);      // global_prefetch_b8 into L2

    if (tid == 0) s_count = 0;
    __syncthreads();

    // ---- phase 1: compact candidates whose disc overlaps this tile ----
    const int w0 = tw * TILE, h0 = th * TILE;
    const float xmax = 1.0f - 2.0f * (w0 + 0.5f) / IMG + r_ndc;
    const float xmin = 1.0f - 2.0f * (w0 + TILE - 1 + 0.5f) / IMG - r_ndc;
    const float ymax = 1.0f - 2.0f * (h0 + 0.5f) / IMG + r_ndc;
    const float ymin = 1.0f - 2.0f * (h0 + TILE - 1 + 0.5f) / IMG - r_ndc;

    const float* pb = pts3D + (size_t)b * NPTS * 3;
    for (int p = tid; p < NPTS; p += 256) {
        float fx = -pb[p * 3 + 0];       // sign flip from torch forward
        float fy = -pb[p * 3 + 1];
        float z  =  pb[p * 3 + 2];
        if (fx >= xmin && fx <= xmax && fy >= ymin && fy <= ymax) {
            int pos = atomicAdd(&s_count, 1);   // ds_add_rtn_u32
            if (pos < MAXC) {
                s_pidx[pos] = p; s_px[pos] = fx; s_py[pos] = fy; s_pz[pos] = z;
            }
        }
    }
    __syncthreads();
    int Nc = s_count; if (Nc > MAXC) Nc = MAXC;

    // ---- phase 2: per-pixel top-K-by-z selection over candidates ----
    const int pxl = tid & 15;
    const int pyl = tid >> 4;
    const float cx = 1.0f - 2.0f * (w0 + pxl + 0.5f) / IMG;
    const float cy = 1.0f - 2.0f * (h0 + pyl + 0.5f) / IMG;

    float zk[KSEL]; int lk[KSEL]; float dk[KSEL];
#pragma unroll
    for (int k = 0; k < KSEL; ++k) { zk[k] = BIGZ; lk[k] = -1; dk[k] = 0.0f; }

    for (int j = 0; j < Nc; ++j) {
        float dx = cx - s_px[j];
        float dy = cy - s_py[j];
        float d2 = dx * dx + dy * dy;
        if (d2 <= r2) {
            float z = s_pz[j];
            if (z < zk[KSEL - 1]) {
                zk[KSEL - 1] = z; lk[KSEL - 1] = j; dk[KSEL - 1] = d2;
#pragma unroll
                for (int k = KSEL - 1; k > 0; --k) {
                    if (zk[k] < zk[k - 1]) {
                        float tz = zk[k]; zk[k] = zk[k - 1]; zk[k - 1] = tz;
                        int   ti = lk[k]; lk[k] = lk[k - 1]; lk[k - 1] = ti;
                        float td = dk[k]; dk[k] = dk[k - 1]; dk[k - 1] = td;
                    }
                }
            }
        }
    }

    // composite weights (TAU = 1): w_k = a_k * prod_{j<k}(1 - a_j)
    float wk[KSEL];
    {
        float T = 1.0f;
#pragma unroll
        for (int k = 0; k < KSEL; ++k) {
            float dist = dk[k] * inv_r2;
            dist = fminf(fmaxf(dist, 0.001f), 1.0f);
            float a = 1.0f - sqrtf(dist);
            a = (zk[k] < BIGZ) ? a : 0.0f;
            wk[k] = a * T;
            T *= (1.0f - a);
        }
    }

    // ---- phase 3: out[64c x 256px] += src_gather[64c x Nc] * W[Nc x 256px] ----
    // wave -> 8 WMMA output tiles: fixed 16-channel band, 8 pixel rows
    const int ct   = wv & 3;             // channel tile 0..3
    const int pgb  = (wv >> 2) * 8;      // pixel-row group base (0 or 8)
    const int cb   = ct * 16;
    const int half = lane >> 4;          // lane half for K/M split
    const int l16  = lane & 15;

    v8f acc[8];
#pragma unroll
    for (int t = 0; t < 8; ++t) acc[t] = (v8f){0,0,0,0,0,0,0,0};

    for (int j0 = 0; j0 < Nc; j0 += 16) {
        // zero + scatter: each pixel thread owns one contiguous 16-float row
        {
            v4f z4 = (v4f){0.0f, 0.0f, 0.0f, 0.0f};
            v4f* row = (v4f*)&s_B[tid * 16];
            row[0] = z4; row[1] = z4; row[2] = z4; row[3] = z4;
        }
#pragma unroll
        for (int k = 0; k < KSEL; ++k) {
            unsigned rel = (unsigned)(lk[k] - j0);
            if (rel < 16u) s_B[tid * 16 + rel] = wk[k];
        }
        // gather A = src[:, cand[j0..j0+16)] : 64x16, 4 elements per thread
#pragma unroll
        for (int i = 0; i < 4; ++i) {
            int e  = tid * 4 + i;
            int c  = e >> 4;
            int kk = e & 15;
            float v = 0.0f;
            if (j0 + kk < Nc) v = srcb[(size_t)c * NPTS + s_pidx[j0 + kk]];
            s_A[c * 16 + kk] = v;
        }
        __syncthreads();

#pragma unroll
        for (int k0 = 0; k0 < 16; k0 += 4) {
            // A fragment 16x4 f32: M = l16, VGPR pair = K {0,1}+2*half (8B aligned)
            v2f af = *(const v2f*)&s_A[(cb + l16) * 16 + k0 + 2 * half];
#pragma unroll
            for (int t = 0; t < 8; ++t) {
                int pix = (pgb + t) * 16 + l16;
                // B fragment 4x16 f32: N = l16, K {0,1}+2*half — contiguous pair
                v2f bf = *(const v2f*)&s_B[pix * 16 + k0 + 2 * half];
                acc[t] = __builtin_amdgcn_wmma_f32_16x16x4_f32(
                    false, af, false, bf, (short)0, acc[t], false, false);
            }
        }
        __syncthreads();
    }

    // ---- phase 4: write D tiles. acc[t][v] -> c = cb + v + 8*half,
    //      h = h0 + (pgb+t), w = w0 + l16
#pragma unroll
    for (int t = 0; t < 8; ++t) {
        int pg = pgb + t;
        float* op = out + (((size_t)b * NCH + cb + 8 * half) * IMG + (h0 + pg)) * IMG
                        + w0 + l16;
#pragma unroll
        for (int v = 0; v < 8; ++v)
            op[(size_t)v * IMG * IMG] = acc[t][v];
    }
}

extern "C" void kernel_launch(void* const* d_in, const int* in_sizes, int n_in,
                              void* d_out, int out_size, void* d_ws, size_t ws_size,
                              hipStream_t stream) {
    const float* pts3D = (const float*)d_in[0];   // [2,2048,3] f32
    const float* src   = (const float*)d_in[1];   // [2,64,2048] f32
    float* out         = (float*)d_out;           // [2,64,128,128] f32
    (void)in_sizes; (void)n_in; (void)out_size; (void)d_ws; (void)ws_size;

    dim3 grid(IMG / TILE, IMG / TILE, 2);         // 8 x 8 x 2 tiles
    dim3 block(256);                               // 8 wave32s
    raster_tile_wmma<<<grid, block, 0, stream>>>(pts3D, src, out);
}